// NodeEdgeProjection_8091718386319
// MI455X (gfx1250) — compile-verified
//
#include <hip/hip_runtime.h>
#include <stdint.h>

// Problem constants (fixed by the reference): B=128, N=128, F=64
#define N_NODES          128
#define FEAT             64
#define EDGES            (N_NODES * (N_NODES - 1))   // 16256
#define ROW_BYTES        (FEAT * 4)                  // 256 B per row
#define CHUNKS_PER_BATCH 8
#define CHUNK_EDGES      (EDGES / CHUNKS_PER_BATCH)  // 2032 = 16 * 127
#define CHUNK_SRC        (N_NODES / CHUNKS_PER_BATCH)// 16 source rows / chunk
#define THREADS          256                          // 8 wave32s

// out[b, e, :] = x[b, e/127, :]   (idx[e] = e/127 analytically, no index buffer)
__global__ __launch_bounds__(THREADS)
void node_edge_gather(const float* __restrict__ x, float* __restrict__ out) {
    __shared__ float tile[CHUNK_SRC * FEAT];          // 16 rows * 256 B = 4 KB

    const unsigned b     = blockIdx.x / CHUNKS_PER_BATCH;
    const unsigned chunk = blockIdx.x % CHUNKS_PER_BATCH;
    const unsigned tid   = threadIdx.x;

    // ---- Phase 1: async DMA 16 source rows (4 KB, contiguous) global -> LDS
    // Each byte of x is read exactly once across the whole grid -> NT load.
    {
        uint64_t g = (uint64_t)(uintptr_t)x
                   + (uint64_t)(b * N_NODES + chunk * CHUNK_SRC) * ROW_BYTES
                   + (uint64_t)tid * 16u;             // one b128 per thread
        unsigned  l = (unsigned)(uintptr_t)tile + tid * 16u; // LDS byte offset
        asm volatile("global_load_async_to_lds_b128 %0, %1, off th:TH_LOAD_NT"
                     :: "v"(l), "v"(g) : "memory");
    }
    asm volatile("s_wait_asynccnt 0" ::: "memory");   // per-wave async loads done
    __syncthreads();                                  // tile visible to all waves

    // ---- Phase 2: stream 2032 edge rows straight from LDS to HBM
    const unsigned wave = tid >> 5;
    const unsigned lane = tid & 31u;
    const unsigned half = lane >> 4;                  // which of the 2 rows/wave
    const unsigned col  = lane & 15u;                 // float4 within 256 B row
    const unsigned k    = wave * 2u + half;           // 0..15, row-in-group id

    const unsigned ldsCol  = (unsigned)(uintptr_t)tile + col * 16u;
    const uint64_t outBase = (uint64_t)(uintptr_t)out
        + (uint64_t)(b * EDGES + chunk * CHUNK_EDGES) * ROW_BYTES
        + (uint64_t)col * 16u;

    #pragma unroll 4
    for (unsigned iter = 0; iter < CHUNK_EDGES / 16u; ++iter) {  // 127 iters
        const unsigned eLocal = iter * 16u + k;       // 0..2031 (exact cover)
        const unsigned sLocal = eLocal / 127u;        // 0..15 -> magic-mul div
        const unsigned l      = ldsCol + sLocal * ROW_BYTES;
        const uint64_t g      = outBase + (uint64_t)eLocal * ROW_BYTES;
        // 512 B per wave per instruction, LDS -> HBM, no VGPR data movement.
        // Output is write-once/stream -> non-temporal, don't pollute L2.
        asm volatile("global_store_async_from_lds_b128 %0, %1, off th:TH_STORE_NT"
                     :: "v"(g), "v"(l) : "memory");
    }
    asm volatile("s_wait_asynccnt 0" ::: "memory");   // drain before endpgm
}

extern "C" void kernel_launch(void* const* d_in, const int* in_sizes, int n_in,
                              void* d_out, int out_size, void* d_ws, size_t ws_size,
                              hipStream_t stream) {
    (void)in_sizes; (void)n_in; (void)out_size; (void)d_ws; (void)ws_size;
    const float* x  = (const float*)d_in[0];
    float*      out = (float*)d_out;

    dim3 grid(128u * CHUNKS_PER_BATCH);   // 128 batches x 8 chunks = 1024 blocks
    dim3 block(THREADS);                  // 256 threads = 8 wave32s
    hipLaunchKernelGGL(node_edge_gather, grid, block, 0, stream, x, out);
}